// MambaBlock_3745211482737
// MI455X (gfx1250) — compile-verified
//
#include <hip/hip_runtime.h>
#include <hip/hip_bf16.h>
#include <math.h>

// Problem constants (match reference)
#define BQ      8
#define LQ      4096
#define DMODEL  256
#define DINNER  1024
#define DSTATE  16
#define DTRANK  16
#define DCONV   4
#define NTOK    (BQ * LQ)           // 32768 tokens
#define XPROJ_N (DTRANK + 2*DSTATE) // 48
#define EPSF    1e-5f

typedef __attribute__((ext_vector_type(2))) float v2f;
typedef __attribute__((ext_vector_type(8))) float v8f;

__device__ __forceinline__ float softplusf(float x) {
  return (x > 20.f) ? x : __logf(1.f + __expf(x));
}
__device__ __forceinline__ float siluf(float x) {
  return x / (1.f + __expf(-x));
}

// ---------------------------------------------------------------------------
// RMSNorm: one wave per token (256 elems -> 8 per lane), shfl reduction.
// ---------------------------------------------------------------------------
__global__ void rmsnorm_kernel(const float* __restrict__ x,
                               const float* __restrict__ w,
                               float* __restrict__ h) {
  int wave = (blockIdx.x * blockDim.x + threadIdx.x) >> 5;
  int lane = threadIdx.x & 31;
  if (wave >= NTOK) return;
  const float* row = x + (size_t)wave * DMODEL;
  float v[8];
  float s = 0.f;
#pragma unroll
  for (int i = 0; i < 8; ++i) {
    v[i] = row[lane + 32 * i];
    s += v[i] * v[i];
  }
#pragma unroll
  for (int m = 1; m < 32; m <<= 1) s += __shfl_xor(s, m, 32);
  float scale = rsqrtf(s * (1.f / DMODEL) + EPSF);
  float* out = h + (size_t)wave * DMODEL;
#pragma unroll
  for (int i = 0; i < 8; ++i)
    out[lane + 32 * i] = v[i] * scale * w[lane + 32 * i];
}

// ---------------------------------------------------------------------------
// FP32 WMMA GEMM:  C[M,N] = A[M,K](lda) * W[N,K]^T  (+ optional residual)
// Each wave computes a 16 x (16*NT) slab. K is processed in chunks of 16:
// all 4+4*NT fragments of a chunk are loaded into DISTINCT registers first,
// then 4*NT WMMAs consume them -- distinct destinations + in-order load
// returns let the scheduler overlap the load clause with the WMMA chain via
// partial s_wait_loadcnt instead of a full wait per WMMA. The A fragment of
// each K-substep feeds NT back-to-back WMMAs (4x A reuse). Weights are
// L2-resident (192MB L2); A streams once.
// Fragment layout per CDNA5 ISA 7.12.2:
//   A 16x4 f32: lane m = lane&15; VGPR v holds K = (lane>>4)*2 + v
//   B  4x16   : lane n = lane&15; VGPR v holds K = (lane>>4)*2 + v
//   C/D 16x16 : VGPR r -> M = r + 8*(lane>>4), N = lane&15
// Wave-uniform bounds guard keeps EXEC all-ones for WMMA.
// Requires: M%16==0, N%(16*NT)==0, K%16==0.
// ---------------------------------------------------------------------------
template <int NT>
__global__ void gemm_f32_wmma(const float* __restrict__ A,
                              const float* __restrict__ W,
                              const float* __restrict__ Res,
                              float* __restrict__ C,
                              int M, int N, int K, int lda, int addRes) {
  int wave = (blockIdx.x * blockDim.x + threadIdx.x) >> 5;
  int lane = threadIdx.x & 31;
  int slabsN = N / (16 * NT);
  int slabs = (M >> 4) * slabsN;
  if (wave >= slabs) return;                  // uniform per wave
  int tm = (wave / slabsN) << 4;
  int tn = (wave % slabsN) * (16 * NT);
  int l15 = lane & 15;
  int hi  = lane >> 4;

  const float* arow = A + (size_t)(tm + l15) * lda + hi * 2;
  const float* brow[NT];
#pragma unroll
  for (int j = 0; j < NT; ++j)
    brow[j] = W + (size_t)(tn + j * 16 + l15) * K + hi * 2;

  v8f acc[NT];
#pragma unroll
  for (int j = 0; j < NT; ++j) acc[j] = {};

  for (int k0 = 0; k0 < K; k0 += 16) {
    v2f a[4];
    v2f b[NT][4];
#pragma unroll
    for (int s = 0; s < 4; ++s) {
      a[s] = *(const v2f*)(arow + k0 + 4 * s);   // global_load_b64
#pragma unroll
      for (int j = 0; j < NT; ++j)
        b[j][s] = *(const v2f*)(brow[j] + k0 + 4 * s);
    }
#pragma unroll
    for (int s = 0; s < 4; ++s)
#pragma unroll
      for (int j = 0; j < NT; ++j)
        acc[j] = __builtin_amdgcn_wmma_f32_16x16x4_f32(
            /*neg_a=*/false, a[s], /*neg_b=*/false, b[j][s],
            /*c_mod=*/(short)0, acc[j], /*reuse_a=*/false, /*reuse_b=*/false);
  }

  int mBase = tm + hi * 8;
#pragma unroll
  for (int j = 0; j < NT; ++j) {
    int nOut = tn + j * 16 + l15;
#pragma unroll
    for (int r = 0; r < 8; ++r) {
      size_t idx = (size_t)(mBase + r) * N + nOut;
      float vv = acc[j][r];
      if (addRes) vv += Res[idx];
      C[idx] = vv;
    }
  }
}

// ---------------------------------------------------------------------------
// Depthwise causal conv1d (taps=4) + bias + SiLU. xc read from xz[:, 0:1024].
// ---------------------------------------------------------------------------
__global__ void conv_silu_kernel(const float* __restrict__ xz,
                                 const float* __restrict__ cw,   // [DINNER,4]
                                 const float* __restrict__ cb,   // [DINNER]
                                 float* __restrict__ xa) {       // [NTOK,DINNER]
  int idx = blockIdx.x * blockDim.x + threadIdx.x;
  if (idx >= NTOK * DINNER) return;
  int c = idx & (DINNER - 1);
  int t = idx >> 10;              // token = b*L + l
  int l = t & (LQ - 1);
  float acc = cb[c];
#pragma unroll
  for (int j = 0; j < DCONV; ++j) {
    int lj = l - (DCONV - 1) + j;
    if (lj >= 0)
      acc += cw[c * DCONV + j] *
             xz[(size_t)(t - (DCONV - 1) + j) * (2 * DINNER) + c];
  }
  xa[(size_t)t * DINNER + c] = siluf(acc);
}

// ---------------------------------------------------------------------------
// Selective scan. One state (d,n) per lane: a wave owns channels (d0, d0+1)
// of one batch, 16 states each -> 4096 concurrent waves,
// y = sum_n h*C via
// 4 shfl_xor steps confined to each 16-lane group.
// Epilogue (y + xa*D) * silu(z) fused; y written into xc half of xz
// (dead after conv) so out_proj reads it with lda = 2048.
// ---------------------------------------------------------------------------
__global__ void scan_kernel(const float* __restrict__ dt_raw,  // [NTOK,DINNER]
                            const float* __restrict__ dt_bias, // [DINNER]
                            const float* __restrict__ xa,      // [NTOK,DINNER]
                            float* __restrict__ xz,            // [NTOK,2*DINNER]
                            const float* __restrict__ xdbl,    // [NTOK,48]
                            const float* __restrict__ A_log,   // [DINNER,16]
                            const float* __restrict__ Dvec) {  // [DINNER]
  int wave = (blockIdx.x * blockDim.x + threadIdx.x) >> 5;
  int lane = threadIdx.x & 31;
  const int wavesPerB = DINNER / 2;           // 512
  if (wave >= BQ * wavesPerB) return;
  int b = wave / wavesPerB;
  int d = ((wave % wavesPerB) << 1) + (lane >> 4);
  int n = lane & 15;

  float Aval = -__expf(A_log[d * DSTATE + n]);
  float Dval = Dvec[d];
  float dtb  = dt_bias[d];
  float h = 0.f;

  size_t row = (size_t)b * LQ;
  for (int l = 0; l < LQ; ++l, ++row) {
    // prefetch the rows we will consume 8 steps ahead (global_prefetch_b8)
    if (l + 8 < LQ) {
      __builtin_prefetch(dt_raw + (row + 8) * DINNER + d, 0, 1);
      __builtin_prefetch(xa + (row + 8) * DINNER + d, 0, 1);
      __builtin_prefetch(xdbl + (row + 8) * XPROJ_N + n, 0, 1);
    }
    float dtr = dt_raw[row * DINNER + d];
    float xv  = xa[row * DINNER + d];
    float Bv  = xdbl[row * XPROJ_N + DTRANK + n];
    float Cv  = xdbl[row * XPROJ_N + DTRANK + DSTATE + n];

    float dt = softplusf(dtr + dtb);
    float dA = __expf(dt * Aval);
    h = fmaf(dA, h, dt * xv * Bv);

    float p = h * Cv;
    p += __shfl_xor(p, 1, 32);
    p += __shfl_xor(p, 2, 32);
    p += __shfl_xor(p, 4, 32);
    p += __shfl_xor(p, 8, 32);                // full sum in all 16 lanes

    if (n == 0) {
      float zv = xz[row * (2 * DINNER) + DINNER + d];
      xz[row * (2 * DINNER) + d] = (p + xv * Dval) * siluf(zv);
    }
  }
}

// ---------------------------------------------------------------------------
// Host-side launch. Workspace partition (all fp32):
//   h      NTOK*256   =  33.5 MB
//   xz     NTOK*2048  = 268.4 MB   (xc half reused for gated y)
//   xa     NTOK*1024  = 134.2 MB
//   xdbl   NTOK*48    =   6.3 MB
//   dtraw  NTOK*1024  = 134.2 MB
//   total ~ 577 MB
// ---------------------------------------------------------------------------
extern "C" void kernel_launch(void* const* d_in, const int* in_sizes, int n_in,
                              void* d_out, int out_size, void* d_ws, size_t ws_size,
                              hipStream_t stream) {
  const float* x         = (const float*)d_in[0];
  const float* norm_w    = (const float*)d_in[1];
  const float* in_proj_w = (const float*)d_in[2];   // [2048, 256]
  const float* conv_w    = (const float*)d_in[3];   // [1024, 1, 4]
  const float* conv_b    = (const float*)d_in[4];
  const float* x_proj_w  = (const float*)d_in[5];   // [48, 1024]
  const float* dt_proj_w = (const float*)d_in[6];   // [1024, 16]
  const float* dt_proj_b = (const float*)d_in[7];
  const float* A_log     = (const float*)d_in[8];   // [1024, 16]
  const float* Dv        = (const float*)d_in[9];
  const float* out_proj_w= (const float*)d_in[10];  // [256, 1024]
  float* out = (float*)d_out;

  char* ws = (char*)d_ws;
  float* h     = (float*)ws;   ws += (size_t)NTOK * DMODEL * 4;
  float* xz    = (float*)ws;   ws += (size_t)NTOK * 2 * DINNER * 4;
  float* xa    = (float*)ws;   ws += (size_t)NTOK * DINNER * 4;
  float* xdbl  = (float*)ws;   ws += (size_t)NTOK * XPROJ_N * 4;
  float* dtraw = (float*)ws;

  const int TB = 256;                    // 8 waves / block

  // 1) RMSNorm: 32768 waves
  rmsnorm_kernel<<<(NTOK * 32) / TB, TB, 0, stream>>>(x, norm_w, h);

  // 2) in_proj: M=32768 N=2048 K=256 -> 65536 slab-waves (16x64 each)
  {
    int slabs = (NTOK / 16) * (2 * DINNER / 64);
    gemm_f32_wmma<4><<<(slabs * 32) / TB, TB, 0, stream>>>(
        h, in_proj_w, nullptr, xz, NTOK, 2 * DINNER, DMODEL, DMODEL, 0);
  }

  // 3) depthwise conv + SiLU: 33.5M threads
  conv_silu_kernel<<<(NTOK * DINNER) / TB, TB, 0, stream>>>(xz, conv_w, conv_b, xa);

  // 4) x_proj: M=32768 N=48 K=1024 (NT=3 slab covers all 48 cols)
  {
    int slabs = (NTOK / 16) * (XPROJ_N / 48);
    gemm_f32_wmma<3><<<(slabs * 32) / TB, TB, 0, stream>>>(
        xa, x_proj_w, nullptr, xdbl, NTOK, XPROJ_N, DINNER, DINNER, 0);
  }

  // 5) dt_proj: M=32768 N=1024 K=16; A = dt slice = xdbl cols [0,16) via lda=48
  {
    int slabs = (NTOK / 16) * (DINNER / 64);
    gemm_f32_wmma<4><<<(slabs * 32) / TB, TB, 0, stream>>>(
        xdbl, dt_proj_w, nullptr, dtraw, NTOK, DINNER, DTRANK, XPROJ_N, 0);
  }

  // 6) selective scan + gate/skip epilogue: 4096 waves, y -> xz[:, 0:1024]
  scan_kernel<<<(BQ * (DINNER / 2) * 32) / TB, TB, 0, stream>>>(
      dtraw, dt_proj_b, xa, xz, xdbl, A_log, Dv);

  // 7) out_proj + residual: M=32768 N=256 K=1024; A = xz (lda=2048)
  {
    int slabs = (NTOK / 16) * (DMODEL / 64);
    gemm_f32_wmma<4><<<(slabs * 32) / TB, TB, 0, stream>>>(
        xz, out_proj_w, x, out, NTOK, DMODEL, DINNER, 2 * DINNER, 1);
  }
}